// XFlatRGBExtractorOp_43258910605672
// MI455X (gfx1250) — compile-verified
//
#include <hip/hip_runtime.h>

// Pure streaming selection kernel for the X-Trans RGB extractor.
// B=16, H=W=1536 (fixed by the reference). 0 FLOPs, ~1.21 GB of traffic
// -> HBM-bound (~52 us floor at 23.3 TB/s). Strategy: one thread per
// float4, b128 loads/stores, non-temporal cache policy on everything.

typedef float v4f __attribute__((ext_vector_type(4)));

static constexpr unsigned W4  = 1536u / 4u;            // vec4s per row
static constexpr unsigned HW4 = (1536u * 1536u) / 4u;  // vec4s per plane

// 6-bit per-row column masks for the 6x6 tile, packed 6 rows * 6 bits.
// R_POS rows: {4},{0,2},{4},{1},{3,5},{1}
static constexpr unsigned long long RPACK =
    (16ull << 0) | (5ull << 6) | (16ull << 12) | (2ull << 18) | (40ull << 24) | (2ull << 30);
// B_POS rows: {1},{3,5},{1},{4},{0,2},{4}
static constexpr unsigned long long BPACK =
    (2ull << 0) | (40ull << 6) | (2ull << 12) | (16ull << 18) | (5ull << 24) | (16ull << 30);

__global__ __launch_bounds__(256) void xtrans_extract_kernel(
    const v4f* __restrict__ green,   // (B,1,H,W)
    const v4f* __restrict__ xtrans,  // (B,3,H,W)
    const v4f* __restrict__ chroma,  // (B,2,H,W)
    v4f* __restrict__ out,           // (B,3,H,W)
    unsigned n4)                     // B*H*W/4
{
    unsigned t = blockIdx.x * blockDim.x + threadIdx.x;
    if (t >= n4) return;

    unsigned b    = t / HW4;          // batch index (const divisor -> mulhi)
    unsigned rem4 = t - b * HW4;      // vec4 index within the (H,W) plane
    unsigned y    = rem4 / W4;        // row
    unsigned x4   = rem4 - y * W4;    // vec4 column

    unsigned r = y % 6u;              // row phase in 6x6 tile
    unsigned g = x4 % 3u;             // vec4 phase (lcm(4,6)=12 -> 3 vec4 phases)

    unsigned rm = (unsigned)(RPACK >> (6u * r)) & 63u;
    unsigned bm = (unsigned)(BPACK >> (6u * r)) & 63u;
    unsigned rn = ((rm | (rm << 6)) >> (4u * g)) & 0xFu;  // 4 mask bits (R)
    unsigned bn = ((bm | (bm << 6)) >> (4u * g)) & 0xFu;  // 4 mask bits (B)

    unsigned x3 = b * (3u * HW4);     // xtrans / out batch base (vec4 units)
    unsigned c2 = b * (2u * HW4);     // chroma batch base

    // Read-once streams: non-temporal b128 loads.
    v4f gr  = __builtin_nontemporal_load(&green[t]);
    v4f xt0 = __builtin_nontemporal_load(&xtrans[x3 + rem4]);
    v4f xt2 = __builtin_nontemporal_load(&xtrans[x3 + 2u * HW4 + rem4]);
    v4f cp0 = __builtin_nontemporal_load(&chroma[c2 + rem4]);
    v4f cp1 = __builtin_nontemporal_load(&chroma[c2 + HW4 + rem4]);

    v4f o0, o2;
#pragma unroll
    for (int i = 0; i < 4; ++i) {
        o0[i] = ((rn >> i) & 1u) ? xt0[i] : cp0[i];
        o2[i] = ((bn >> i) & 1u) ? xt2[i] : cp1[i];
    }

    // Write-once stream: non-temporal b128 stores.
    __builtin_nontemporal_store(o0, &out[x3 + rem4]);             // ch0 (R select)
    __builtin_nontemporal_store(gr, &out[x3 + HW4 + rem4]);       // ch1 (green copy)
    __builtin_nontemporal_store(o2, &out[x3 + 2u * HW4 + rem4]);  // ch2 (B select)
}

extern "C" void kernel_launch(void* const* d_in, const int* in_sizes, int n_in,
                              void* d_out, int out_size, void* d_ws, size_t ws_size,
                              hipStream_t stream) {
    (void)n_in; (void)out_size; (void)d_ws; (void)ws_size;
    const v4f* green  = (const v4f*)d_in[0];  // float32 (B,1,H,W)
    const v4f* xtrans = (const v4f*)d_in[1];  // float32 (B,3,H,W)
    const v4f* chroma = (const v4f*)d_in[2];  // float32 (B,2,H,W)
    v4f* out = (v4f*)d_out;                   // float32 (B,3,H,W)

    unsigned n4 = (unsigned)(in_sizes[0] / 4); // B*H*W/4 vec4 work items
    dim3 block(256);
    dim3 grid((n4 + block.x - 1) / block.x);
    xtrans_extract_kernel<<<grid, block, 0, stream>>>(green, xtrans, chroma, out, n4);
}